// AdvancedTokenRouter_19559281066377
// MI455X (gfx1250) — compile-verified
//
#include <hip/hip_runtime.h>
#include <hip/hip_bf16.h>

typedef __attribute__((ext_vector_type(2))) float v2f;
typedef __attribute__((ext_vector_type(8))) float v8f;

// Monotonic float -> uint mapping: a > b  <=>  key(a) > key(b)
__device__ __forceinline__ unsigned f2key(float f) {
    unsigned u = __float_as_uint(f);
    return (u & 0x80000000u) ? ~u : (u | 0x80000000u);
}

// ---------------------------------------------------------------------------
// logits[E][N] = x[N][D] @ W[E][D]^T + b   via V_WMMA_F32_16X16X4_F32
// Block = 256 threads = 8 waves. Each wave: 16 tokens x 64 experts.
// A frag (x): lane r=lane&15 -> row t0+r; kh=2*(lane>>4) -> K pair {kh, kh+1}
// B frag (W): lane r -> expert row; same K pair. D frag: expert = lane&15,
// token = v + 8*(lane>>4), v = 0..7 (contiguous 8-float store per lane).
// ---------------------------------------------------------------------------
__global__ __launch_bounds__(256) void router_gemm_wmma(
    const float* __restrict__ x, const float* __restrict__ W,
    const float* __restrict__ bias, float* __restrict__ logits,
    int N, int D, int E)
{
    const int lane = threadIdx.x & 31;
    const int wave = threadIdx.x >> 5;
    const int t0 = (blockIdx.x * 8 + wave) * 16;
    if (t0 >= N) return;                 // uniform per wave: EXEC stays all-1s

    const int r  = lane & 15;
    const int kh = (lane >> 4) << 1;     // 0 or 2

    const float* xrow = x + (size_t)(t0 + r) * D + kh;
    const float* w0 = W + (size_t)(0 * 16 + r) * D + kh;
    const float* w1 = W + (size_t)(1 * 16 + r) * D + kh;
    const float* w2 = W + (size_t)(2 * 16 + r) * D + kh;
    const float* w3 = W + (size_t)(3 * 16 + r) * D + kh;

    v8f acc0 = {}, acc1 = {}, acc2 = {}, acc3 = {};

#pragma unroll 2
    for (int k = 0; k < D; k += 4) {
        v2f a  = *(const v2f*)(xrow + k);
        v2f b0 = *(const v2f*)(w0 + k);
        v2f b1 = *(const v2f*)(w1 + k);
        v2f b2 = *(const v2f*)(w2 + k);
        v2f b3 = *(const v2f*)(w3 + k);
        acc0 = __builtin_amdgcn_wmma_f32_16x16x4_f32(false, a, false, b0, (short)0, acc0, false, false);
        acc1 = __builtin_amdgcn_wmma_f32_16x16x4_f32(false, a, false, b1, (short)0, acc1, false, false);
        acc2 = __builtin_amdgcn_wmma_f32_16x16x4_f32(false, a, false, b2, (short)0, acc2, false, false);
        acc3 = __builtin_amdgcn_wmma_f32_16x16x4_f32(false, a, false, b3, (short)0, acc3, false, false);
    }

    const int trow = t0 + ((lane >> 4) << 3);   // token of VGPR 0 for this lane
    v8f accs[4] = {acc0, acc1, acc2, acc3};
#pragma unroll
    for (int eb = 0; eb < 4; ++eb) {
        const int e = eb * 16 + r;
        const float bv = bias[e];
        float* dst = logits + (size_t)e * N + trow;
        v8f a = accs[eb];
#pragma unroll
        for (int v = 0; v < 8; ++v) dst[v] = a[v] + bv;
    }
}

// ---------------------------------------------------------------------------
// Per-expert radix-select of the k-th largest logit (threshold key tau[e]).
// One block per expert; 4 MSB-first passes over the expert's contiguous row.
// Also zeroes counts[e] for the next kernel (stream order = sync).
// ---------------------------------------------------------------------------
__global__ __launch_bounds__(256) void expert_select(
    const float* __restrict__ logits, unsigned* __restrict__ tau,
    float* __restrict__ counts, int N, int k)
{
    const int e = blockIdx.x;
    if (threadIdx.x == 0) counts[e] = 0.0f;

    __shared__ unsigned hist[256];
    __shared__ unsigned s_bin, s_rem;

    const float* col = logits + (size_t)e * N;
    unsigned prefix = 0, pmask = 0;
    unsigned kk = (unsigned)k;

    for (int pass = 0; pass < 4; ++pass) {
        const int shift = 24 - 8 * pass;
        hist[threadIdx.x] = 0;
        __syncthreads();
        for (int t = threadIdx.x; t < N; t += 256) {
            unsigned key = f2key(col[t]);
            if ((key & pmask) == prefix)
                atomicAdd(&hist[(key >> shift) & 255u], 1u);
        }
        __syncthreads();
        if (threadIdx.x == 0) {
            unsigned cum = 0, rem = kk;
            int bsel = 0;
            for (int b = 255; b >= 0; --b) {
                unsigned c = hist[b];
                if (cum + c >= kk) { bsel = b; rem = kk - cum; break; }
                cum += c;
            }
            s_bin = (unsigned)bsel;
            s_rem = rem;
        }
        __syncthreads();
        prefix |= s_bin << shift;
        pmask  |= 0xFFu << shift;
        kk = s_rem;
    }
    if (threadIdx.x == 0) tau[e] = prefix;
}

// ---------------------------------------------------------------------------
// Per-token: best candidate logit, smallest expert achieving it, selected
// flag, and per-expert counts. logits is [E][N] so reads are fully coalesced.
// ---------------------------------------------------------------------------
__global__ __launch_bounds__(256) void token_assign(
    const float* __restrict__ logits, const unsigned* __restrict__ tau,
    float* __restrict__ out_best, float* __restrict__ out_exp,
    float* __restrict__ out_sel, float* __restrict__ counts, int N, int E)
{
    const int t = blockIdx.x * blockDim.x + threadIdx.x;
    if (t >= N) return;

    float best = -3.402823466e+38f;
    int bexp = -1;
    for (int e = 0; e < E; ++e) {
        float v = logits[(size_t)e * N + t];
        if (f2key(v) >= tau[e]) {
            if (v > best) { best = v; bexp = e; }   // strict > : smallest e wins ties
        }
    }
    const bool sel = (bexp >= 0);
    out_best[t] = sel ? best : 0.0f;
    out_exp[t]  = sel ? (float)bexp : -1.0f;
    out_sel[t]  = sel ? 1.0f : 0.0f;
    if (sel) atomicAdd(&counts[bexp], 1.0f);
}

// ---------------------------------------------------------------------------
// load_balance_loss = mean((counts - mean)^2) / (mean + 1e-9)
// ---------------------------------------------------------------------------
__global__ __launch_bounds__(64) void lb_loss(
    const float* __restrict__ counts, float* __restrict__ out_loss, int E)
{
    __shared__ float sh[64];
    const int i = threadIdx.x;
    sh[i] = (i < E) ? counts[i] : 0.0f;
    __syncthreads();
    if (i == 0) {
        float s = 0.0f;
        for (int j = 0; j < E; ++j) s += sh[j];
        const float mean = s / (float)E;
        float ss = 0.0f;
        for (int j = 0; j < E; ++j) { float d = sh[j] - mean; ss += d * d; }
        *out_loss = (ss / (float)E) / (mean + 1e-9f);
    }
}

extern "C" void kernel_launch(void* const* d_in, const int* in_sizes, int n_in,
                              void* d_out, int out_size, void* d_ws, size_t ws_size,
                              hipStream_t stream) {
    const float* x = (const float*)d_in[0];
    const float* W = (const float*)d_in[1];
    const float* b = (const float*)d_in[2];

    const int E = in_sizes[2];                 // 64
    const int D = in_sizes[1] / E;             // 1024
    const int N = (int)((long long)in_sizes[0] / D);  // 32768

    int k = (int)((double)N / (double)(E > 1 ? E : 1) * 1.2);
    if (k < 1) k = 1;
    if (k > N) k = N;

    float*    logits = (float*)d_ws;                                   // E*N f32
    unsigned* tau    = (unsigned*)((char*)d_ws + (size_t)E * N * sizeof(float));
    float*    counts = (float*)(tau + E);

    float* out_best = (float*)d_out;
    float* out_exp  = out_best + N;
    float* out_sel  = out_exp + N;
    float* out_loss = out_sel + N;

    router_gemm_wmma<<<dim3((N + 127) / 128), 256, 0, stream>>>(x, W, b, logits, N, D, E);
    expert_select<<<dim3(E), 256, 0, stream>>>(logits, tau, counts, N, k);
    token_assign<<<dim3((N + 255) / 256), 256, 0, stream>>>(logits, tau, out_best, out_exp, out_sel, counts, N, E);
    lb_loss<<<dim3(1), 64, 0, stream>>>(counts, out_loss, E);
}